// MultiHeadAttention_11081015623957
// MI455X (gfx1250) — compile-verified
//
#include <hip/hip_runtime.h>
#include <hip/hip_bf16.h>

#define Bq  8
#define Hh  12
#define LKk 4096
#define DMm 1024
#define DHd 512
#define HDe (Hh * DHd)   // 6144

typedef __attribute__((ext_vector_type(2))) float v2f;
typedef __attribute__((ext_vector_type(8))) float v8f;

// ---------------------------------------------------------------------------
// Generic small GEMM via V_WMMA_F32_16X16X4_F32.
//   D[m, n0+n] = sum_k A[m&7, k] * W[n0+n, k] + bias[n0+n],  m = 0..7 valid
// A is 8 x K row-major (rows 8..15 of the WMMA tile read duplicated rows;
// their C rows are simply never written). W is N x K row-major.
// One wave per block; block b covers N-columns [16b, 16b+16).
// ---------------------------------------------------------------------------
__global__ __launch_bounds__(32) void wmma_gemm_f32(
    const float* __restrict__ A, const float* __restrict__ W,
    const float* __restrict__ bias, float* __restrict__ D,
    int K, int ldD) {
  const int lane  = threadIdx.x;          // 0..31, wave32
  const int n0    = blockIdx.x * 16;
  const int mn    = lane & 15;            // M index for A, N index for B
  const int khalf = (lane >> 4) * 2;      // 0 for lanes 0-15, 2 for 16-31
  const float* Arow = A + (size_t)(mn & 7) * K;
  const float* Wrow = W + (size_t)(n0 + mn) * K;
  v8f c = {};
  for (int kk = 0; kk < K; kk += 4) {
    v2f a = *(const v2f*)(Arow + kk + khalf);
    v2f b = *(const v2f*)(Wrow + kk + khalf);
    c = __builtin_amdgcn_wmma_f32_16x16x4_f32(
        /*neg_a=*/false, a, /*neg_b=*/false, b,
        /*c_mod=*/(short)0, c, /*reuse_a=*/false, /*reuse_b=*/false);
  }
  if (lane < 16) {
    float bv = bias[n0 + lane];
#pragma unroll
    for (int r = 0; r < 8; ++r)           // C VGPR r = row M=r for lanes 0-15
      D[(size_t)r * ldD + n0 + lane] = c[r] + bv;
  }
}

// ---------------------------------------------------------------------------
// scores + mask + softmax -> attn (written directly into mul_attn output).
// One block per (b,h); 32 waves; each wave streams 128 key rows (coalesced
// float4), lane-reduces the 512-dot with __shfl_xor (wave32).
// global_prefetch_b8 issued 2 rows ahead (each lane covers its 64B slice).
// ---------------------------------------------------------------------------
__global__ __launch_bounds__(1024) void scores_softmax_kernel(
    const float* __restrict__ qh, const float* __restrict__ k_s,
    const int* __restrict__ mask, float* __restrict__ attn) {
  __shared__ float qlds[DHd];
  __shared__ float sc[LKk];
  __shared__ float red[1024];
  const int bh = blockIdx.x;
  const int t = threadIdx.x;
  const int lane = t & 31, wave = t >> 5;       // 32 waves
  if (t < DHd) qlds[t] = qh[(size_t)(bh / Hh) * HDe + (bh % Hh) * DHd + t];
  __syncthreads();

  const float scale = 0.04419417382415922f;     // 512^-0.5
  const float* kbase = k_s + (size_t)bh * LKk * DHd;
  const int* mrow = mask + (size_t)bh * LKk;
  const float4* q4 = (const float4*)qlds;
  for (int i = 0; i < 128; ++i) {
    const int k = wave * 128 + i;
    const float4* krow = (const float4*)(kbase + (size_t)k * DHd);
    if (i + 2 < 128) {                          // prefetch 2 rows ahead
      __builtin_prefetch(kbase + (size_t)(k + 2) * DHd + lane * 16, 0, 3);
    }
    float acc = 0.f;
#pragma unroll
    for (int j = 0; j < 4; ++j) {
      float4 kv = krow[lane + 32 * j];
      float4 qv = q4[lane + 32 * j];
      acc += kv.x * qv.x + kv.y * qv.y + kv.z * qv.z + kv.w * qv.w;
    }
#pragma unroll
    for (int off = 16; off > 0; off >>= 1) acc += __shfl_xor(acc, off, 32);
    if (lane == 0) sc[k] = mrow[k] ? -1000000.0f : acc * scale;
  }
  __syncthreads();

  float m = -3.4e38f;
  for (int k = t; k < LKk; k += 1024) m = fmaxf(m, sc[k]);
  red[t] = m; __syncthreads();
  for (int s = 512; s > 0; s >>= 1) {
    if (t < s) red[t] = fmaxf(red[t], red[t + s]);
    __syncthreads();
  }
  const float mx = red[0];
  __syncthreads();

  float lsum = 0.f;
  for (int k = t; k < LKk; k += 1024) {
    float e = __expf(sc[k] - mx);
    sc[k] = e;
    lsum += e;
  }
  red[t] = lsum; __syncthreads();
  for (int s = 512; s > 0; s >>= 1) {
    if (t < s) red[t] += red[t + s];
    __syncthreads();
  }
  const float inv = 1.f / red[0];
  float* arow = attn + (size_t)bh * LKk;
  for (int k = t; k < LKk; k += 1024) arow[k] = sc[k] * inv;
}

// ---------------------------------------------------------------------------
// context partial: out[(bh*nsplit+ks)*512 + d] = sum over keys slice of
// attn[k]*v[k][d]. 128 threads; the v_s stream uses double-buffered
// GLOBAL_LOAD_ASYNC_TO_LDS_B128 (ASYNCcnt) — each thread consumes exactly
// the 16B it DMA'd, so a per-wave s_wait_asynccnt suffices (no barrier).
// ---------------------------------------------------------------------------
__global__ __launch_bounds__(128) void ctx_partial_kernel(
    const float* __restrict__ attn, const float* __restrict__ v_s,
    float* __restrict__ outp, int kper, int nsplit) {
  __shared__ float alds[LKk];
  __shared__ float vbuf[2][DHd];
  const int bh = blockIdx.x, ks = blockIdx.y, t = threadIdx.x;
  const int k0 = ks * kper;
  const float* arow = attn + (size_t)bh * LKk + k0;
  for (int k = t; k < kper; k += 128) alds[k] = arow[k];
  __syncthreads();

  const float* vbase = v_s + ((size_t)bh * LKk + k0) * DHd;
  const unsigned l0 = (unsigned)(size_t)&vbuf[0][4 * t];
  const unsigned l1 = (unsigned)(size_t)&vbuf[1][4 * t];
  // prologue: async-load row 0 into buffer 0
  asm volatile("global_load_async_to_lds_b128 %0, %1, off"
               :: "v"(l0), "v"((unsigned long long)(size_t)(vbase + 4 * t))
               : "memory");
  float4 acc = {0.f, 0.f, 0.f, 0.f};
  for (int k = 0; k < kper; ++k) {
    if (k + 1 < kper) {
      const unsigned ln = ((k + 1) & 1) ? l1 : l0;
      const unsigned long long gn =
          (unsigned long long)(size_t)(vbase + (size_t)(k + 1) * DHd + 4 * t);
      asm volatile("global_load_async_to_lds_b128 %0, %1, off"
                   :: "v"(ln), "v"(gn) : "memory");
      asm volatile("s_wait_asynccnt 0x1" ::: "memory");   // row k landed
    } else {
      asm volatile("s_wait_asynccnt 0x0" ::: "memory");
    }
    const float a = alds[k];
    const float4 v = ((const float4*)vbuf[k & 1])[t];
    acc.x += a * v.x; acc.y += a * v.y; acc.z += a * v.z; acc.w += a * v.w;
  }
  ((float4*)(outp + ((size_t)(bh * nsplit + ks)) * DHd))[t] = acc;
}

__global__ __launch_bounds__(128) void ctx_reduce_kernel(
    const float* __restrict__ ctxp, float* __restrict__ ctx, int nsplit) {
  const int bh = blockIdx.x, t = threadIdx.x;
  const float4* p = (const float4*)(ctxp + (size_t)bh * nsplit * DHd);
  float4 acc = {0.f, 0.f, 0.f, 0.f};
  for (int s = 0; s < nsplit; ++s) {
    float4 v = p[(size_t)s * 128 + t];
    acc.x += v.x; acc.y += v.y; acc.z += v.z; acc.w += v.w;
  }
  ((float4*)(ctx + (size_t)bh * DHd))[t] = acc;
}

__global__ __launch_bounds__(128) void head_weights_kernel(
    const float* __restrict__ ctx, const float* __restrict__ lw_w,
    const float* __restrict__ lw_b, float* __restrict__ wout) {
  __shared__ float red[128];
  const int bh = blockIdx.x, t = threadIdx.x;
  float4 c = ((const float4*)(ctx + (size_t)bh * DHd))[t];
  float4 w = ((const float4*)lw_w)[t];
  red[t] = c.x * w.x + c.y * w.y + c.z * w.z + c.w * w.w;
  __syncthreads();
  for (int s = 64; s > 0; s >>= 1) {
    if (t < s) red[t] += red[t + s];
    __syncthreads();
  }
  if (t == 0) wout[bh] = red[0] + lw_b[0];
}

__global__ __launch_bounds__(32) void argmax_sel_kernel(
    const float* __restrict__ w, int* __restrict__ sel) {
  const int b = threadIdx.x;
  if (b < Bq) {
    float best = w[b * Hh];
    int bi = 0;
    for (int h = 1; h < Hh; ++h) {
      float v = w[b * Hh + h];
      if (v > best) { best = v; bi = h; }   // strict > : first max, like jnp.argmax
    }
    sel[b] = bi;
  }
}

__global__ __launch_bounds__(256) void gather_kernel(
    const float* __restrict__ attn, const float* __restrict__ ctx,
    const int* __restrict__ sel, float* __restrict__ ret_attn,
    float* __restrict__ osel) {
  const int b = blockIdx.x, t = threadIdx.x;
  const int s = sel[b];
  const float* arow = attn + ((size_t)b * Hh + s) * LKk;
  float* rrow = ret_attn + (size_t)b * LKk;
  for (int k = t; k < LKk; k += 256) rrow[k] = arow[k];
  const float* crow = ctx + ((size_t)b * Hh + s) * DHd;
  float* orow = osel + (size_t)b * DHd;
  for (int d = t; d < DHd; d += 256) orow[d] = crow[d];
}

// ---------------------------------------------------------------------------
extern "C" void kernel_launch(void* const* d_in, const int* in_sizes, int n_in,
                              void* d_out, int out_size, void* d_ws, size_t ws_size,
                              hipStream_t stream) {
  const float* Q    = (const float*)d_in[0];
  const float* k_s  = (const float*)d_in[1];
  const float* v_s  = (const float*)d_in[2];
  const int*   mask = (const int*)d_in[3];
  /* d_in[4] = base_flag : unused by reference */
  const float* WQw  = (const float*)d_in[5];
  const float* WQb  = (const float*)d_in[6];
  const float* lww  = (const float*)d_in[7];
  const float* lwb  = (const float*)d_in[8];
  const float* lf1w = (const float*)d_in[9];
  const float* lf1b = (const float*)d_in[10];
  const float* lf2w = (const float*)d_in[11];
  const float* lf2b = (const float*)d_in[12];

  // d_out layout: output | ret_attn | mul_attn | context
  float* out   = (float*)d_out;                  // 8*1024
  float* retat = out + Bq * DMm;                 // 8*4096
  float* mul   = retat + Bq * LKk;               // 8*12*4096
  float* ctx   = mul + (size_t)Bq * Hh * LKk;    // 8*12*512

  // Workspace layout (floats)
  float* wsf = (float*)d_ws;
  float* qh = wsf;                               // 49152
  size_t off = (size_t)Bq * HDe;

  const size_t need_big = (off + 96ull * 8 * DHd + 96 + 4096 + 4096 + 8) * 4ull;
  const int KS = (ws_size >= need_big) ? 8 : 1;  // ws_size constant -> deterministic

  float* ctxp = wsf + off;
  if (KS > 1) off += 96ull * KS * DHd;           // partial context buffer
  float* wts  = wsf + off; off += 96;
  float* osel = wsf + off; off += (size_t)Bq * DHd;
  float* out1 = wsf + off; off += (size_t)Bq * DHd;
  int*   sel  = (int*)(wsf + off);

  // 1) q projection: qh[b, h*512+d]  (WMMA f32 16x16x4)
  wmma_gemm_f32<<<HDe / 16, 32, 0, stream>>>(Q, WQw, WQb, qh, DMm, HDe);

  // 2) scores + mask + softmax -> mul_attn (805 MB k_s stream)
  scores_softmax_kernel<<<Bq * Hh, 1024, 0, stream>>>(qh, k_s, mask, mul);

  // 3) context = attn @ v  (805 MB v_s stream, async-to-LDS double buffer)
  if (KS > 1) {
    ctx_partial_kernel<<<dim3(Bq * Hh, KS), 128, 0, stream>>>(mul, v_s, ctxp, LKk / KS, KS);
    ctx_reduce_kernel<<<Bq * Hh, 128, 0, stream>>>(ctxp, ctx, KS);
  } else {
    ctx_partial_kernel<<<dim3(Bq * Hh, 1), 128, 0, stream>>>(mul, v_s, ctx, LKk, 1);
  }

  // 4) per-head gating weight + argmax head selection
  head_weights_kernel<<<Bq * Hh, 128, 0, stream>>>(ctx, lww, lwb, wts);
  argmax_sel_kernel<<<1, 32, 0, stream>>>(wts, sel);

  // 5) gather ret_attn and the selected context rows
  gather_kernel<<<Bq, 256, 0, stream>>>(mul, ctx, sel, retat, osel);

  // 6) lf1 (8x512 @ 512x512) then lf2 (8x512 @ 1024x512) via WMMA f32
  wmma_gemm_f32<<<DHd / 16, 32, 0, stream>>>(osel, lf1w, lf1b, out1, DHd, DHd);
  wmma_gemm_f32<<<DMm / 16, 32, 0, stream>>>(out1, lf2w, lf2b, out, DHd, DMm);
}